// GatedSelfAttention_49254684950677
// MI455X (gfx1250) — compile-verified
//
#include <hip/hip_runtime.h>
#include <stdint.h>

typedef __attribute__((ext_vector_type(2))) float v2f;
typedef __attribute__((ext_vector_type(8))) float v8f;

#define B_    4
#define S_    4096
#define D_    128
#define QTW   16            // query rows per wave
#define WAVES 4             // waves per workgroup
#define KT    16            // key-tile (keys per iteration)
#define LDST  132           // padded LDS row stride in floats (16B aligned, bank-spread)
#define TILEF (KT * LDST)   // floats per staged tile

__device__ __forceinline__ void async_b128(uint32_t lds_addr, const float* g) {
  asm volatile("global_load_async_to_lds_b128 %0, %1, off"
               :: "v"(lds_addr), "v"((unsigned long long)(uintptr_t)g)
               : "memory");
}
__device__ __forceinline__ void wait_async0() {
  asm volatile("s_wait_asynccnt 0" ::: "memory");
}

__device__ __forceinline__ v8f v8f_zero() {
  v8f z;
#pragma unroll
  for (int i = 0; i < 8; ++i) z[i] = 0.0f;
  return z;
}

// Issue one K/V key-tile (16 keys x 128 d) into LDS via async-to-LDS b128.
// 128 lanes x 4 iters x 16B = 8KB per matrix.
__device__ __forceinline__ void issue_tile(float* sK, float* sV,
                                           const float* Kb, const float* Vb,
                                           int k0, int tid) {
#pragma unroll
  for (int i = 0; i < 4; ++i) {
    int f   = tid + 128 * i;      // float4 index 0..511
    int row = f >> 5;             // 0..15 (key row)
    int c4  = (f & 31) << 2;      // float column 0..124
    async_b128((uint32_t)(uintptr_t)(sK + row * LDST + c4),
               Kb + (size_t)(k0 + row) * D_ + c4);
    async_b128((uint32_t)(uintptr_t)(sV + row * LDST + c4),
               Vb + (size_t)(k0 + row) * D_ + c4);
  }
}

__global__ __launch_bounds__(128, 1)
void gsa_flash_kernel(const float* __restrict__ Q, const float* __restrict__ K,
                      const float* __restrict__ V, float* __restrict__ O) {
  __shared__ float smem[2 * 2 * TILEF];   // double-buffered K,V tiles

  const int tid  = threadIdx.x;
  const int lane = tid & 31;
  const int wave = tid >> 5;
  const int l15  = lane & 15;
  const int h    = lane >> 4;             // lane half (0/1)

  const int blocksPerBatch = S_ / (QTW * WAVES);            // 64
  const int b   = blockIdx.x / blocksPerBatch;
  const int q0  = (blockIdx.x % blocksPerBatch) * (QTW * WAVES) + wave * QTW;

  const float* Qb = Q + (size_t)b * S_ * D_;
  const float* Kb = K + (size_t)b * S_ * D_;
  const float* Vb = V + (size_t)b * S_ * D_;
  float*       Ob = O + (size_t)b * S_ * D_;

  // ---- Q fragments held in registers for the whole sweep (64 VGPRs/lane) ----
  // B-layout chunk c: lane half h reads Q[q=l15][4c+2h .. 4c+2h+1]
  v2f qf[32];
#pragma unroll
  for (int c = 0; c < 32; ++c)
    qf[c] = *(const v2f*)(Qb + (size_t)(q0 + l15) * D_ + 4 * c + 2 * h);

  // ---- output accumulators: 8 C-tiles (16 q x 16 d each) = 64 VGPRs ----
  v8f acc[8];
#pragma unroll
  for (int i = 0; i < 8; ++i) acc[i] = v8f_zero();
  float m  = -1e30f;   // running row max (per query, lanes q and q+16)
  float ls = 0.0f;     // running row sum

  // Prologue: stage tile 0 into buffer 0
  issue_tile(smem, smem + TILEF, Kb, Vb, 0, tid);

  const int NT = S_ / KT;  // 256 key tiles
  for (int kt = 0; kt < NT; ++kt) {
    wait_async0();       // this lane's tile-kt data is in LDS
    __syncthreads();     // everyone's tile-kt is in LDS; everyone done with kt-1

    if (kt + 1 < NT) {   // prefetch next tile into the other buffer
      float* base = smem + ((kt + 1) & 1) * (2 * TILEF);
      issue_tile(base, base + TILEF, Kb, Vb, (kt + 1) * KT, tid);
    }

    const float* sK = smem + (kt & 1) * (2 * TILEF);
    const float* sV = sK + TILEF;

    // ---- GEMM1: S^T (16 keys x 16 queries) = K_tile(16x128) * Q^T ----
    // Two independent accumulator chains (even/odd chunks) so the scheduler
    // can overlap one chain's WMMA issue with the other's DS-load waits.
    v8f st0 = v8f_zero();
    v8f st1 = v8f_zero();
#pragma unroll
    for (int c = 0; c < 16; ++c) {
      v2f a0 = *(const v2f*)(sK + l15 * LDST + 4 * (2 * c)     + 2 * h);
      v2f a1 = *(const v2f*)(sK + l15 * LDST + 4 * (2 * c + 1) + 2 * h);
      st0 = __builtin_amdgcn_wmma_f32_16x16x4_f32(
              false, a0, false, qf[2 * c],     (short)0, st0, false, false);
      st1 = __builtin_amdgcn_wmma_f32_16x16x4_f32(
              false, a1, false, qf[2 * c + 1], (short)0, st1, false, false);
    }
    v8f st;
#pragma unroll
    for (int r = 0; r < 8; ++r) st[r] = st0[r] + st1[r];
    // st[r] @ lane l  ->  score(key = r + 8h, query = l15)

    // ---- online softmax (row = query = l15, replicated across halves) ----
    float tm = st[0];
#pragma unroll
    for (int r = 1; r < 8; ++r) tm = fmaxf(tm, st[r]);
    tm = fmaxf(tm, __shfl_xor(tm, 16, 32));
    float mn = fmaxf(m, tm);
    float sc = __expf(m - mn);
    m = mn;
    ls *= sc;

    float p[8];
    float tsum = 0.0f;
#pragma unroll
    for (int r = 0; r < 8; ++r) { p[r] = __expf(st[r] - mn); tsum += p[r]; }
    tsum += __shfl_xor(tsum, 16, 32);
    ls += tsum;

    // rescale accumulators: C-tile reg r holds query (r + 8h) at this lane
#pragma unroll
    for (int r = 0; r < 8; ++r) {
      float fr = __shfl(sc, r + 8 * h, 32);
#pragma unroll
      for (int db = 0; db < 8; ++db) acc[db][r] *= fr;
    }

    // ---- GEMM2: acc(16q x 128d) += P(16q x 16k) * V_tile(16k x 128d) ----
    // 8 independent accumulator chains (one per 16-wide d block).
    // A frag for K-chunk kc built from p[] (S^T C-layout) via shuffles.
#pragma unroll
    for (int kc = 0; kc < 4; ++kc) {
      const int kx0 = 4 * kc,     kx1 = 4 * kc + 2;   // a.x key for h=0 / h=1
      const int ky0 = 4 * kc + 1, ky1 = 4 * kc + 3;   // a.y key for h=0 / h=1
      float ax0 = __shfl(p[kx0 & 7], l15 + ((kx0 & 8) ? 16 : 0), 32);
      float ax1 = __shfl(p[kx1 & 7], l15 + ((kx1 & 8) ? 16 : 0), 32);
      float ay0 = __shfl(p[ky0 & 7], l15 + ((ky0 & 8) ? 16 : 0), 32);
      float ay1 = __shfl(p[ky1 & 7], l15 + ((ky1 & 8) ? 16 : 0), 32);
      v2f a; a.x = h ? ax1 : ax0; a.y = h ? ay1 : ay0;

      const float* vrow = sV + (4 * kc + 2 * h) * LDST;  // keys k0+2h, k0+2h+1
#pragma unroll
      for (int db = 0; db < 8; ++db) {
        v2f bb;
        bb.x = vrow[db * 16 + l15];
        bb.y = vrow[LDST + db * 16 + l15];
        acc[db] = __builtin_amdgcn_wmma_f32_16x16x4_f32(
                     false, a, false, bb, (short)0, acc[db], false, false);
      }
    }
  }

  // ---- epilogue: divide by row sum, gate with V, store ----
#pragma unroll
  for (int r = 0; r < 8; ++r) {
    float lr  = __shfl(ls, r + 8 * h, 32);
    float inv = 1.0f / lr;
    int   q   = q0 + r + 8 * h;
#pragma unroll
    for (int db = 0; db < 8; ++db) {
      int d = db * 16 + l15;
      float gate = Vb[(size_t)q * D_ + d];
      Ob[(size_t)q * D_ + d] = acc[db][r] * inv * gate;
    }
  }
}

extern "C" void kernel_launch(void* const* d_in, const int* in_sizes, int n_in,
                              void* d_out, int out_size, void* d_ws, size_t ws_size,
                              hipStream_t stream) {
  (void)in_sizes; (void)n_in; (void)out_size; (void)d_ws; (void)ws_size;
  const float* Q = (const float*)d_in[0];
  const float* K = (const float*)d_in[1];
  const float* V = (const float*)d_in[2];
  float*       O = (float*)d_out;

  dim3 grid(B_ * (S_ / (QTW * WAVES)));  // 4 * 64 = 256 workgroups
  dim3 block(QTW * WAVES * 2);           // 128 threads = 4 waves
  gsa_flash_kernel<<<grid, block, 0, stream>>>(Q, K, V, O);
}